// ARAPLoss_32083405701678
// MI455X (gfx1250) — compile-verified
//
#include <hip/hip_runtime.h>

typedef __attribute__((ext_vector_type(2))) float v2f;
typedef __attribute__((ext_vector_type(8))) float v8f;

// ---------------------------------------------------------------------------
// Exact f32 wave(32)-wide sum using V_WMMA_F32_16X16X4_F32.
// A = ones(16x4) => D[m,n] = column sums of B; layout-agnostic since A == 1.
// Lane l ends holding colsum(l & 15); xor-reduce over {8,4,2,1} -> full total.
// ---------------------------------------------------------------------------
__device__ __forceinline__ float wave_sum_wmma(float x) {
    v2f a; a.x = 1.0f; a.y = 1.0f;
    v2f b; b.x = x;    b.y = 0.0f;
    v8f c = {};
    v8f d = __builtin_amdgcn_wmma_f32_16x16x4_f32(false, a, false, b,
                                                  (short)0, c, false, false);
    float s = d[0];
    s += __shfl_xor(s, 8, 32);
    s += __shfl_xor(s, 4, 32);
    s += __shfl_xor(s, 2, 32);
    s += __shfl_xor(s, 1, 32);
    return s;
}

// Sum of singular values of 3x3 S via closed-form eigenvalues of B = S^T S.
__device__ __forceinline__ float nuclear_norm3(
    float s00, float s01, float s02,
    float s10, float s11, float s12,
    float s20, float s21, float s22) {
    float b00 = s00*s00 + s10*s10 + s20*s20;
    float b11 = s01*s01 + s11*s11 + s21*s21;
    float b22 = s02*s02 + s12*s12 + s22*s22;
    float b01 = s00*s01 + s10*s11 + s20*s21;
    float b02 = s00*s02 + s10*s12 + s20*s22;
    float b12 = s01*s02 + s11*s12 + s21*s22;

    float q   = (b00 + b11 + b22) * (1.0f / 3.0f);
    float m00 = b00 - q, m11 = b11 - q, m22 = b22 - q;
    float p2  = m00*m00 + m11*m11 + m22*m22
              + 2.0f * (b01*b01 + b02*b02 + b12*b12);
    float p   = sqrtf(p2 * (1.0f / 6.0f));

    float det = m00 * (m11*m22 - b12*b12)
              - b01 * (b01*m22 - b12*b02)
              + b02 * (b01*b12 - m11*b02);
    float denom = 2.0f * p * p * p;
    float r = (denom > 1e-30f) ? (det / denom) : 0.0f;
    r = fminf(1.0f, fmaxf(-1.0f, r));

    float phi = acosf(r) * (1.0f / 3.0f);
    float e1 = q + 2.0f * p * cosf(phi);
    float e3 = q + 2.0f * p * cosf(phi + 2.09439510239f); // + 2*pi/3
    float e2 = 3.0f * q - e1 - e3;

    return sqrtf(fmaxf(e1, 0.0f)) + sqrtf(fmaxf(e2, 0.0f)) + sqrtf(fmaxf(e3, 0.0f));
}

// Per-vertex ARAP contribution:  sum_k w(|d|^2+|r|^2) - 2*||S||_* .
template <typename PF, typename PI>
__device__ __forceinline__ float arap_vertex(const float* __restrict__ Vd,
                                             PF rrow, PF wrow, PI irow, int K) {
    float s00 = 0.f, s01 = 0.f, s02 = 0.f;
    float s10 = 0.f, s11 = 0.f, s12 = 0.f;
    float s20 = 0.f, s21 = 0.f, s22 = 0.f;
    float a0  = 0.f;
    #pragma unroll
    for (int k = 0; k < K; ++k) {
        const int es = irow[2*k + 0];
        const int et = irow[2*k + 1];
        const float dx = Vd[3*et + 0] - Vd[3*es + 0];
        const float dy = Vd[3*et + 1] - Vd[3*es + 1];
        const float dz = Vd[3*et + 2] - Vd[3*es + 2];
        const float w  = wrow[k];
        const float rx = rrow[3*k + 0];
        const float ry = rrow[3*k + 1];
        const float rz = rrow[3*k + 2];
        const float wrx = w * rx, wry = w * ry, wrz = w * rz;
        s00 = fmaf(wrx, dx, s00); s01 = fmaf(wrx, dy, s01); s02 = fmaf(wrx, dz, s02);
        s10 = fmaf(wry, dx, s10); s11 = fmaf(wry, dy, s11); s12 = fmaf(wry, dz, s12);
        s20 = fmaf(wrz, dx, s20); s21 = fmaf(wrz, dy, s21); s22 = fmaf(wrz, dz, s22);
        a0  = fmaf(w, dx*dx + dy*dy + dz*dz + rx*rx + ry*ry + rz*rz, a0);
    }
    const float nuc = nuclear_norm3(s00, s01, s02, s10, s11, s12, s20, s21, s22);
    return a0 - 2.0f * nuc;
}

__device__ __forceinline__ void block_store_partial(float wsum, float* partials) {
    __shared__ float sm[32];
    const int lane = threadIdx.x & 31;
    const int wid  = threadIdx.x >> 5;
    if (lane == 0) sm[wid] = wsum;
    __syncthreads();
    if (threadIdx.x == 0) {
        float t = 0.0f;
        const int nw = blockDim.x >> 5;
        for (int i = 0; i < nw; ++i) t += sm[i];
        partials[blockIdx.x] = t;
    }
}

// ---------------------------------------------------------------------------
// Async global->LDS staging (gfx1250):  each wave copies `bytes` (multiple of
// 256, 16B-aligned base) from global to LDS with coalesced 512B-per-issue
// transfers that never touch VGPRs. Tracked by ASYNCcnt.
// ---------------------------------------------------------------------------
__device__ __forceinline__ void stage_async(unsigned int lds, unsigned long long g,
                                            int bytes) {
    const int lane = threadIdx.x & 31;
    const int b128_bytes = bytes & ~511;          // full 512B wave rounds
    for (int off = lane * 16; off < b128_bytes; off += 512) {
        unsigned int       l = lds + (unsigned int)off;
        unsigned long long a = g + (unsigned long long)off;
        asm volatile("global_load_async_to_lds_b128 %0, %1, off"
                     :: "v"(l), "v"(a) : "memory");
    }
    if (bytes & 256) {                            // 256B tail as one b64 round
        unsigned int       l = lds + (unsigned int)b128_bytes + (unsigned int)(lane << 3);
        unsigned long long a = g + (unsigned long long)b128_bytes + (unsigned long long)(lane << 3);
        asm volatile("global_load_async_to_lds_b64 %0, %1, off"
                     :: "v"(l), "v"(a) : "memory");
    }
}

// ---------------------------------------------------------------------------
// Staged kernel: K == 18, N % 128 == 0.  128 threads = 4 wave32s; each wave
// stages its 32-vertex tile (rest 6912B | wts 2304B | idx 4608B = 13824B)
// into its private LDS slice, waits on ASYNCcnt, then consumes via ds_load.
// ---------------------------------------------------------------------------
#define TILE_BYTES 13824
#define REST_BYTES 6912
#define WTS_BYTES  2304
#define IDX_BYTES  4608

__global__ void __launch_bounds__(128)
arap_staged_kernel(const float* __restrict__ Vd,
                   const float* __restrict__ rest,
                   const float* __restrict__ wts,
                   const int*   __restrict__ idx,
                   float* __restrict__ partials, int N) {
    __shared__ __align__(16) char smem[4][TILE_BYTES];

    const int lane = threadIdx.x & 31;
    const int wid  = threadIdx.x >> 5;
    const int n0   = blockIdx.x * 128 + wid * 32;   // wave's first vertex
    const int n    = n0 + lane;                     // this lane's vertex

    const unsigned int lds_base = (unsigned int)(uintptr_t)&smem[wid][0];

    // Wave-coalesced async staging of the three element streams.
    stage_async(lds_base,                            (unsigned long long)(uintptr_t)(rest + (size_t)n0 * 54u), REST_BYTES);
    stage_async(lds_base + REST_BYTES,               (unsigned long long)(uintptr_t)(wts  + (size_t)n0 * 18u), WTS_BYTES);
    stage_async(lds_base + REST_BYTES + WTS_BYTES,   (unsigned long long)(uintptr_t)(idx  + (size_t)n0 * 36u), IDX_BYTES);
    asm volatile("s_wait_asynccnt 0x0" ::: "memory");

    const float* rrow = (const float*)&smem[wid][lane * 216];
    const float* wrow = (const float*)&smem[wid][REST_BYTES + lane * 72];
    const int*   irow = (const int*)  &smem[wid][REST_BYTES + WTS_BYTES + lane * 144];

    float acc = (n < N) ? arap_vertex(Vd, rrow, wrow, irow, 18) : 0.0f;

    const float wsum = wave_sum_wmma(acc);          // EXEC all-ones here
    block_store_partial(wsum, partials);
}

// ---------------------------------------------------------------------------
// Generic fallback: direct global loads, any N / K.
// ---------------------------------------------------------------------------
__global__ void __launch_bounds__(256)
arap_direct_kernel(const float* __restrict__ Vd,
                   const float* __restrict__ rest,
                   const float* __restrict__ wts,
                   const int*   __restrict__ idx,
                   float* __restrict__ partials, int N, int K) {
    float acc = 0.0f;
    const int stride = gridDim.x * blockDim.x;
    for (int n = blockIdx.x * blockDim.x + threadIdx.x; n < N; n += stride) {
        const float* rrow = rest + (size_t)n * (size_t)K * 3u;
        const float* wrow = wts  + (size_t)n * (size_t)K;
        const int*   irow = idx  + (size_t)n * (size_t)K * 2u;
        acc += arap_vertex(Vd, rrow, wrow, irow, K);
    }
    const float wsum = wave_sum_wmma(acc);
    block_store_partial(wsum, partials);
}

__global__ void __launch_bounds__(256)
arap_finish_kernel(const float* __restrict__ partials,
                   float* __restrict__ out, int nparts) {
    float acc = 0.0f;
    for (int i = threadIdx.x; i < nparts; i += blockDim.x) acc += partials[i];

    const float wsum = wave_sum_wmma(acc);

    __shared__ float sm[32];
    const int lane = threadIdx.x & 31;
    const int wid  = threadIdx.x >> 5;
    if (lane == 0) sm[wid] = wsum;
    __syncthreads();
    if (threadIdx.x == 0) {
        float t = 0.0f;
        const int nw = blockDim.x >> 5;
        for (int i = 0; i < nw; ++i) t += sm[i];
        out[0] = t;
    }
}

extern "C" void kernel_launch(void* const* d_in, const int* in_sizes, int n_in,
                              void* d_out, int out_size, void* d_ws, size_t ws_size,
                              hipStream_t stream) {
    const float* Vd   = (const float*)d_in[0];   // (N,3)   f32
    const float* rest = (const float*)d_in[1];   // (N,K,3) f32
    const float* wts  = (const float*)d_in[2];   // (N,K)   f32
    const int*   idx  = (const int*)  d_in[3];   // (N,K,2) i32

    const int N = in_sizes[0] / 3;
    const int K = (N > 0) ? (in_sizes[2] / N) : 0;

    float* partials = (float*)d_ws;

    int nblocks;
    if (K == 18 && N % 128 == 0 && (size_t)(N / 128) * sizeof(float) <= ws_size) {
        nblocks = N / 128;                       // 2048 for the 512x512 grid
        arap_staged_kernel<<<nblocks, 128, 0, stream>>>(
            Vd, rest, wts, idx, partials, N);
    } else {
        nblocks = 1024;
        arap_direct_kernel<<<nblocks, 256, 0, stream>>>(
            Vd, rest, wts, idx, partials, N, K);
    }
    arap_finish_kernel<<<1, 256, 0, stream>>>(partials, (float*)d_out, nblocks);
}